// ExpressionHierarchyEncoder_22359599743551
// MI455X (gfx1250) — compile-verified
//
#include <hip/hip_runtime.h>
#include <stdint.h>

// Problem constants (fixed by the reference)
#define B_    8
#define S_    8192
#define D_    1024
#define L_    32
#define SCALEF 0.15f

#define SCAN_THREADS 256
#define CHUNK (S_ / SCAN_THREADS)      // 32 tokens per thread

#define POSB  128                      // positions per gather block
#define GATHER_THREADS 256             // 256 threads * float4 = 1024 dims

typedef __attribute__((ext_vector_type(4))) float v4f;

__device__ __forceinline__ int bracket_delta(int t) {
    int o = (t == 40) | (t == 91) | (t == 123);
    int c = (t == 41) | (t == 93) | (t == 125);
    return o - c;
}

// ---------------------------------------------------------------------------
// Kernel 1: parallel saturating-depth scan.
// clip-add functions f(x)=min(max(x+a,lo),hi) compose associatively, so each
// thread composes its 32-token chunk, one thread folds the 256 chunk
// functions to get each chunk's starting level, then each thread replays its
// chunk with the exact reference recurrence (bit-identical levels).
// ---------------------------------------------------------------------------
__global__ __launch_bounds__(SCAN_THREADS)
void ehe_levels_kernel(const int* __restrict__ tok,
                       unsigned char* __restrict__ levels)
{
    const int tid  = threadIdx.x;
    const int row  = blockIdx.x;                // 0..B_-1
    const int base = row * S_ + tid * CHUNK;

    // Compose chunk function (identity = (0, -BIG, +BIG))
    int a = 0, lo = -(1 << 20), hi = (1 << 20);
    #pragma unroll 8
    for (int i = 0; i < CHUNK; ++i) {
        int d = bracket_delta(tok[base + i]);
        a  += d;
        lo  = max(lo + d, 0);
        hi  = min(max(hi + d, 0), L_ - 1);
    }

    __shared__ int sa[SCAN_THREADS], slo[SCAN_THREADS], shi[SCAN_THREADS];
    __shared__ int sstart[SCAN_THREADS];
    sa[tid] = a; slo[tid] = lo; shi[tid] = hi;
    __syncthreads();

    if (tid == 0) {
        int lvl = 0;
        for (int t = 0; t < SCAN_THREADS; ++t) {
            sstart[t] = lvl;
            lvl = min(max(lvl + sa[t], slo[t]), shi[t]);
        }
    }
    __syncthreads();

    // Replay chunk exactly like the reference step
    int lvl = sstart[tid];
    #pragma unroll 8
    for (int i = 0; i < CHUNK; ++i) {
        int d = bracket_delta(tok[base + i]);
        lvl = min(max(lvl + d, 0), L_ - 1);
        levels[base + i] = (unsigned char)lvl;
    }
}

// ---------------------------------------------------------------------------
// Kernel 2: gather + scale, streaming 256MB of output.
//  - async-stage the whole 32x1024 f32 table (128KB) into LDS (ASYNCcnt path)
//  - pre-scale by 0.15 once in LDS
//  - per position: uniform level from LDS, ds_load_b128, NT global_store_b128
// ---------------------------------------------------------------------------
__global__ __launch_bounds__(GATHER_THREADS)
void ehe_gather_kernel(const float* __restrict__ emb,
                       const unsigned char* __restrict__ levels,
                       float* __restrict__ out)
{
    extern __shared__ float semb[];                 // [L_*D_] + POSB ints
    int* slvl = (int*)(semb + L_ * D_);

    const int  tid     = threadIdx.x;
    const long basePos = (long)blockIdx.x * POSB;

    // Stage 128KB table: 8192 x b128 transfers = 32 per thread, async to LDS.
    // Dynamic LDS starts at byte offset 0 (no static LDS in this kernel).
    {
        const char* g = (const char*)emb;
        #pragma unroll
        for (int i = 0; i < (L_ * D_ * 4) / (GATHER_THREADS * 16); ++i) {
            unsigned    ldsOff = (unsigned)((i * GATHER_THREADS + tid) * 16);
            const char* gp     = g + (size_t)(i * GATHER_THREADS + tid) * 16;
            asm volatile("global_load_async_to_lds_b128 %0, %1, off"
                         :: "v"(ldsOff), "v"(gp)
                         : "memory");
        }
    }

    // Overlap: fetch this block's levels while the DMA is in flight.
    if (tid < POSB) slvl[tid] = levels[basePos + tid];

    asm volatile("s_wait_asynccnt 0" ::: "memory");
    __syncthreads();

    // Scale once in LDS (same f32 result as scaling each output element).
    #pragma unroll 4
    for (int i = tid; i < L_ * D_; i += GATHER_THREADS)
        semb[i] *= SCALEF;
    __syncthreads();

    // Stream the output: one position per iteration, 1024 floats per block.
    float* outp = out + basePos * (long)D_;
    for (int p = 0; p < POSB; ++p) {
        const int lvl = slvl[p];                       // uniform broadcast
        v4f v = ((const v4f*)(semb + lvl * D_))[tid];  // ds_load_b128
        __builtin_nontemporal_store(v, ((v4f*)outp) + tid);
        outp += D_;
    }
}

// ---------------------------------------------------------------------------
extern "C" void kernel_launch(void* const* d_in, const int* in_sizes, int n_in,
                              void* d_out, int out_size, void* d_ws, size_t ws_size,
                              hipStream_t stream) {
    (void)in_sizes; (void)n_in; (void)out_size; (void)ws_size;

    const int*   tok = (const int*)d_in[0];       // [8, 8192] int32
    const float* emb = (const float*)d_in[1];     // [32, 1024] f32
    float*       out = (float*)d_out;             // [8, 8192, 1024] f32
    unsigned char* levels = (unsigned char*)d_ws; // 64KB scratch

    ehe_levels_kernel<<<B_, SCAN_THREADS, 0, stream>>>(tok, levels);

    const size_t dynsz = (size_t)L_ * D_ * sizeof(float) + POSB * sizeof(int);
    hipFuncSetAttribute((const void*)ehe_gather_kernel,
                        hipFuncAttributeMaxDynamicSharedMemorySize, (int)dynsz);

    ehe_gather_kernel<<<(B_ * S_) / POSB, GATHER_THREADS, dynsz, stream>>>(
        emb, levels, out);
}